// AutonomousAI_38182259262002
// MI455X (gfx1250) — compile-verified
//
#include <hip/hip_runtime.h>
#include <hip/hip_bf16.h>
#include <math.h>

typedef __attribute__((ext_vector_type(2))) float v2f;
typedef __attribute__((ext_vector_type(8))) float v8f;

#define ACT_NONE 0
#define ACT_GELU 1

__device__ __forceinline__ float gelu_exact(float x) {
    // exact GELU: 0.5*x*(1+erf(x/sqrt(2)))
    return 0.5f * x * (1.0f + erff(x * 0.7071067811865476f));
}

__device__ __forceinline__ float sigmoidf_(float x) {
    return 1.0f / (1.0f + expf(-x));
}

// ---------------------------------------------------------------------------
// fp32 WMMA GEMM: C[m,n] = act( sum_k (A[m,k] (+A2[m,k])) * W[n,k] + bias[n] )
// V_WMMA_F32_16X16X4_F32; each wave computes a 16x64 strip (4 N-tiles) so the
// A fragment is reused 4x per k-step. The K loop is software-pipelined: the
// fragment loads for step k0+4 are issued before the 4 WMMAs of step k0, so
// global-load latency overlaps the matrix ops instead of serializing them.
// Requires M%16==0, N%64==0, K%4==0, K>=4 (true for every GEMM in this model).
// ---------------------------------------------------------------------------
template<bool HAS_A2, int ACT>
__global__ void __launch_bounds__(256)
gemm_wmma_f32(const float* __restrict__ A, int lda,
              const float* __restrict__ A2,      // elementwise add (same lda) iff HAS_A2
              const float* __restrict__ W, int ldw,
              const float* __restrict__ bias,    // may be null
              float* __restrict__ C, int ldc,
              int M, int N, int K)
{
    const int tilesM  = M >> 4;
    const int groupsN = N >> 6;                  // N / 64
    const int tiles   = tilesM * groupsN;
    const int waveId  = (blockIdx.x * blockDim.x + threadIdx.x) >> 5;
    if (waveId >= tiles) return;                 // wave-uniform: EXEC all-1s for WMMA

    const int m0   = (waveId % tilesM) << 4;
    const int n0   = (waveId / tilesM) << 6;
    const int lane = threadIdx.x & 31;
    const int half = lane >> 4;                  // 0: K pair {0,1}, 1: K pair {2,3}
    const int lq   = lane & 15;

    // A-matrix 16x4 layout: lanes 0-15 hold M rows; v0/v1 = K=2h, K=2h+1
    const float* __restrict__ arow  = A + (m0 + lq) * lda + 2 * half;
    const float* __restrict__ arow2 = HAS_A2 ? (A2 + (m0 + lq) * lda + 2 * half) : nullptr;
    // B-matrix 4x16 layout: lane n holds B[2h+{0,1}, n] = W[n0+n, k0+2h+{0,1}]
    const float* __restrict__ wrow0 = W + (n0 + lq) * ldw + 2 * half;
    const float* __restrict__ wrow1 = wrow0 + 16 * ldw;
    const float* __restrict__ wrow2 = wrow1 + 16 * ldw;
    const float* __restrict__ wrow3 = wrow2 + 16 * ldw;

    v8f c0 = {0.f,0.f,0.f,0.f,0.f,0.f,0.f,0.f};
    v8f c1 = c0, c2 = c0, c3 = c0;

    // ---- prologue: load fragments for k0 = 0 ----
    v2f a, b0, b1, b2, b3;
    a.x  = arow[0];  a.y  = arow[1];
    if (HAS_A2) { a.x += arow2[0]; a.y += arow2[1]; }
    b0.x = wrow0[0]; b0.y = wrow0[1];
    b1.x = wrow1[0]; b1.y = wrow1[1];
    b2.x = wrow2[0]; b2.y = wrow2[1];
    b3.x = wrow3[0]; b3.y = wrow3[1];

    // ---- steady state: prefetch k0+4, then WMMA on k0 ----
    #pragma unroll 2
    for (int k0 = 0; k0 < K - 4; k0 += 4) {
        const int kn = k0 + 4;
        v2f an, b0n, b1n, b2n, b3n;
        an.x  = arow[kn];  an.y  = arow[kn + 1];
        if (HAS_A2) { an.x += arow2[kn]; an.y += arow2[kn + 1]; }
        b0n.x = wrow0[kn]; b0n.y = wrow0[kn + 1];
        b1n.x = wrow1[kn]; b1n.y = wrow1[kn + 1];
        b2n.x = wrow2[kn]; b2n.y = wrow2[kn + 1];
        b3n.x = wrow3[kn]; b3n.y = wrow3[kn + 1];

        c0 = __builtin_amdgcn_wmma_f32_16x16x4_f32(false, a, false, b0, (short)0, c0, false, false);
        c1 = __builtin_amdgcn_wmma_f32_16x16x4_f32(false, a, false, b1, (short)0, c1, false, false);
        c2 = __builtin_amdgcn_wmma_f32_16x16x4_f32(false, a, false, b2, (short)0, c2, false, false);
        c3 = __builtin_amdgcn_wmma_f32_16x16x4_f32(false, a, false, b3, (short)0, c3, false, false);

        a = an; b0 = b0n; b1 = b1n; b2 = b2n; b3 = b3n;
    }

    // ---- epilogue: last k-step ----
    c0 = __builtin_amdgcn_wmma_f32_16x16x4_f32(false, a, false, b0, (short)0, c0, false, false);
    c1 = __builtin_amdgcn_wmma_f32_16x16x4_f32(false, a, false, b1, (short)0, c1, false, false);
    c2 = __builtin_amdgcn_wmma_f32_16x16x4_f32(false, a, false, b2, (short)0, c2, false, false);
    c3 = __builtin_amdgcn_wmma_f32_16x16x4_f32(false, a, false, b3, (short)0, c3, false, false);

    v8f cacc[4] = {c0, c1, c2, c3};
    #pragma unroll
    for (int t = 0; t < 4; ++t) {
        const int col = n0 + 16 * t + lq;
        const float bv = bias ? bias[col] : 0.0f;
        #pragma unroll
        for (int r = 0; r < 8; ++r) {
            const int row = m0 + r + 8 * half;   // D layout: VGPR r -> M=r / M=r+8
            float v = cacc[t][r] + bv;
            if (ACT == ACT_GELU) v = gelu_exact(v);
            C[row * ldc + col] = v;
        }
    }
}

// ---------------------------------------------------------------------------
// Copy action (32x512) into X[:, 512:1024] (X is 32x1024)
// ---------------------------------------------------------------------------
__global__ void concat_action_kernel(const float* __restrict__ action,
                                     float* __restrict__ X)
{
    int idx = blockIdx.x * blockDim.x + threadIdx.x;
    if (idx >= 32 * 512) return;
    int b = idx >> 9, d = idx & 511;
    X[b * 1024 + 512 + d] = action[idx];
}

// ---------------------------------------------------------------------------
// GRU cell with h0 == 0:  gh = bhh (broadcast)
//   r = sigmoid(gi[:, :D]   + bhh[:D])
//   u = sigmoid(gi[:, D:2D] + bhh[D:2D])
//   n = tanh   (gi[:, 2D:]  + r * bhh[2D:])
//   h = (1-u)*n
// gi already contains + gru_bih (added as GEMM bias).
// ---------------------------------------------------------------------------
__global__ void gru_kernel(const float* __restrict__ gi,   // 32x1536
                           const float* __restrict__ bhh,  // 1536
                           float* __restrict__ h)          // 32x512
{
    int idx = blockIdx.x * blockDim.x + threadIdx.x;
    if (idx >= 32 * 512) return;
    int b = idx >> 9, d = idx & 511;
    const float* g = gi + b * 1536;
    float r = sigmoidf_(g[d]        + bhh[d]);
    float u = sigmoidf_(g[512 + d]  + bhh[512 + d]);
    float n = tanhf   (g[1024 + d] + r * bhh[1024 + d]);
    h[idx] = (1.0f - u) * n;
}

// ---------------------------------------------------------------------------
// Fused score/graph kernel. One block per (i,j) in 64x64.
//   graph[i,j] = mean_b sigmoid( sum_k gelu(Cc[b,k]+Ea[i,k]+Eb[j,k]+b1[k]) * w2[k] + b2 )
// Stages (Ea[i]+Eb[j]+b1) and w2 in LDS; 8 waves each own 4 batch rows.
// ---------------------------------------------------------------------------
__global__ void __launch_bounds__(256)
score_graph_kernel(const float* __restrict__ Ea,   // 64x1024
                   const float* __restrict__ Eb,   // 64x1024
                   const float* __restrict__ Cc,   // 32x1024
                   const float* __restrict__ b1,   // 1024
                   const float* __restrict__ w2,   // 1024
                   const float* __restrict__ b2,   // 1
                   float* __restrict__ graph)      // 64x64
{
    const int i = blockIdx.x;
    const int j = blockIdx.y;
    const int t = threadIdx.x;          // 256 threads = 8 waves
    const int wave = t >> 5;
    const int lane = t & 31;

    __shared__ float sb[1024];
    __shared__ float sw[1024];
    __shared__ float red[8];

    const float* ea = Ea + i * 1024;
    const float* eb = Eb + j * 1024;
    for (int k = t; k < 1024; k += 256) {
        sb[k] = ea[k] + eb[k] + b1[k];
        sw[k] = w2[k];
    }
    __syncthreads();

    const float b2v = b2[0];
    float acc = 0.0f;                    // valid in lane 0 of each wave
    for (int b = wave; b < 32; b += 8) {
        const float* cc = Cc + b * 1024;
        float p = 0.0f;
        #pragma unroll 4
        for (int k = lane; k < 1024; k += 32) {
            p += gelu_exact(sb[k] + cc[k]) * sw[k];
        }
        #pragma unroll
        for (int off = 16; off > 0; off >>= 1)
            p += __shfl_down(p, off, 32);
        if (lane == 0) acc += sigmoidf_(p + b2v);
    }
    if (lane == 0) red[wave] = acc;
    __syncthreads();
    if (t == 0) {
        float s = 0.0f;
        #pragma unroll
        for (int wv = 0; wv < 8; ++wv) s += red[wv];
        graph[i * 64 + j] = s * (1.0f / 32.0f);
    }
}

// ---------------------------------------------------------------------------
// Host launcher
// ---------------------------------------------------------------------------
static void launch_gemm(const float* A, int lda, const float* A2,
                        const float* W, int ldw, const float* bias,
                        float* C, int ldc, int M, int N, int K, int act,
                        hipStream_t stream)
{
    int tiles  = (M / 16) * (N / 64);
    int blocks = (tiles + 7) / 8;        // 8 waves (256 threads) per block
    if (A2) {
        gemm_wmma_f32<true, ACT_NONE><<<blocks, 256, 0, stream>>>(
            A, lda, A2, W, ldw, bias, C, ldc, M, N, K);
    } else if (act == ACT_GELU) {
        gemm_wmma_f32<false, ACT_GELU><<<blocks, 256, 0, stream>>>(
            A, lda, nullptr, W, ldw, bias, C, ldc, M, N, K);
    } else {
        gemm_wmma_f32<false, ACT_NONE><<<blocks, 256, 0, stream>>>(
            A, lda, nullptr, W, ldw, bias, C, ldc, M, N, K);
    }
}

extern "C" void kernel_launch(void* const* d_in, const int* in_sizes, int n_in,
                              void* d_out, int out_size, void* d_ws, size_t ws_size,
                              hipStream_t stream) {
    (void)in_sizes; (void)n_in; (void)out_size; (void)ws_size;

    const float* obs     = (const float*)d_in[0];   // 32x768
    const float* action  = (const float*)d_in[1];   // 32x512
    const float* embed   = (const float*)d_in[2];   // 64x512
    const float* sc_w1   = (const float*)d_in[3];   // 1024x1536
    const float* sc_b1   = (const float*)d_in[4];   // 1024
    const float* sc_w2   = (const float*)d_in[5];   // 1x1024
    const float* sc_b2   = (const float*)d_in[6];   // 1
    const float* enc_w1  = (const float*)d_in[7];   // 1024x768
    const float* enc_b1  = (const float*)d_in[8];   // 1024
    const float* enc_w2  = (const float*)d_in[9];   // 512x1024
    const float* enc_b2  = (const float*)d_in[10];  // 512
    const float* gru_wih = (const float*)d_in[11];  // 1536x1024
    /* gru_whh (d_in[12]) unused: h0 == 0 */
    const float* gru_bih = (const float*)d_in[13];  // 1536
    const float* gru_bhh = (const float*)d_in[14];  // 1536
    const float* dec_w   = (const float*)d_in[15];  // 768x512
    const float* dec_b   = (const float*)d_in[16];  // 768
    const float* rs_w1   = (const float*)d_in[17];  // 512x512
    const float* rs_b1   = (const float*)d_in[18];  // 512
    const float* rs_w2   = (const float*)d_in[19];  // 512x512
    const float* rs_b2   = (const float*)d_in[20];  // 512

    float* out   = (float*)d_out;
    float* H     = out;                  // 32x512  = 16384
    float* PRED  = out + 16384;          // 32x768  = 24576
    float* GRAPH = out + 40960;          // 64x64   = 4096
    float* REAS  = out + 45056;          // 32x512  = 16384

    float* ws = (float*)d_ws;
    float* T1 = ws;             // 32x1024
    float* X  = T1 + 32768;     // 32x1024 (concat [z, action])
    float* GI = X  + 32768;     // 32x1536
    float* EA = GI + 49152;     // 64x1024
    float* EB = EA + 65536;     // 64x1024
    float* CC = EB + 65536;     // 32x1024
    float* TR = CC + 32768;     // 32x512

    // Encoder
    launch_gemm(obs, 768, nullptr, enc_w1, 768, enc_b1, T1, 1024, 32, 1024, 768, ACT_GELU, stream);
    launch_gemm(T1, 1024, nullptr, enc_w2, 1024, enc_b2, X,  1024, 32, 512, 1024, ACT_NONE, stream); // z -> X[:, :512]
    concat_action_kernel<<<(32 * 512 + 255) / 256, 256, 0, stream>>>(action, X);

    // GRU (h0 == 0)
    launch_gemm(X, 1024, nullptr, gru_wih, 1024, gru_bih, GI, 1536, 32, 1536, 1024, ACT_NONE, stream);
    gru_kernel<<<(32 * 512 + 255) / 256, 256, 0, stream>>>(GI, gru_bhh, H);

    // Decoder prediction
    launch_gemm(H, 512, nullptr, dec_w, 512, dec_b, PRED, 768, 32, 768, 512, ACT_NONE, stream);

    // Score projections: Wa/Wb/Wc are column slices of sc_w1 (row stride 1536)
    launch_gemm(embed, 512, nullptr, sc_w1,        1536, nullptr, EA, 1024, 64, 1024, 512, ACT_NONE, stream);
    launch_gemm(embed, 512, nullptr, sc_w1 + 512,  1536, nullptr, EB, 1024, 64, 1024, 512, ACT_NONE, stream);
    launch_gemm(H,     512, action,  sc_w1 + 1024, 1536, nullptr, CC, 1024, 32, 1024, 512, ACT_NONE, stream);

    // Fused hid/scores/graph (dominant cost: ~134M exact-GELU evals, fully fused)
    score_graph_kernel<<<dim3(64, 64), 256, 0, stream>>>(EA, EB, CC, sc_b1, sc_w2,
                                                         sc_b2, GRAPH);

    // Reasoning head
    launch_gemm(H,  512, nullptr, rs_w1, 512, rs_b1, TR,   512, 32, 512, 512, ACT_GELU, stream);
    launch_gemm(TR, 512, nullptr, rs_w2, 512, rs_b2, REAS, 512, 32, 512, 512, ACT_NONE, stream);
}